// ForwardKinematicQuat_47622597378599
// MI455X (gfx1250) — compile-verified
//
#include <hip/hip_runtime.h>
#include <math.h>

#define NJ   65
#define BLK  128
#define FK_EPS 1e-5f

// ---- gfx1250 async global<->LDS helpers (ASYNCcnt path) -------------------
// Generic pointers to LDS carry the LDS byte offset in their low 32 bits
// (shared aperture occupies addr[63:32]), so a trunc gives the DS address.
__device__ __forceinline__ void async_ld_dw(void* lds_p, const void* g_base,
                                            unsigned byte_off) {
  unsigned lds_addr = (unsigned)(size_t)lds_p;
  asm volatile("global_load_async_to_lds_b32 %0, %1, %2"
               :: "v"(lds_addr), "v"(byte_off), "s"(g_base)
               : "memory");
}

__device__ __forceinline__ void async_st_dw(void* g_base, const void* lds_p,
                                            unsigned byte_off) {
  unsigned lds_addr = (unsigned)(size_t)lds_p;
  asm volatile("global_store_async_from_lds_b32 %0, %1, %2"
               :: "v"(byte_off), "v"(lds_addr), "s"(g_base)
               : "memory");
}

__device__ __forceinline__ void wait_async0() {
#if __has_builtin(__builtin_amdgcn_s_wait_asynccnt)
  __builtin_amdgcn_s_wait_asynccnt(0);
#else
  asm volatile("s_wait_asynccnt 0x0" ::: "memory");
#endif
}

// ---- kernel ----------------------------------------------------------------
__global__ __launch_bounds__(BLK) void ForwardKinematicQuat_kernel(
    const float* __restrict__ joff,    // [NJ,3] joint_offsets
    const float* __restrict__ rootp,   // [3]    root_position
    const float* __restrict__ wgt,     // [NJ,4] weight (local quats, x,y,z,w)
    const int*   __restrict__ par,     // [NJ]   joint_parents (int32)
    float*       __restrict__ out)     // [NJ,3] world positions
{
  __shared__ float s_w[NJ * 4];
  __shared__ float s_off[NJ * 3];
  __shared__ int   s_par[NJ];
  __shared__ float s_root[4];
  __shared__ float s_ori[NJ * 4];
  __shared__ float s_pos[NJ * 3];
  __shared__ int   s_done[NJ];

  const int tid = threadIdx.x;

  // -- stage all inputs into LDS via the async DMA path (one HBM pass) ------
  for (int k = tid; k < NJ * 4; k += BLK) async_ld_dw(&s_w[k],   wgt,  (unsigned)(k * 4));
  for (int k = tid; k < NJ * 3; k += BLK) async_ld_dw(&s_off[k], joff, (unsigned)(k * 4));
  for (int k = tid; k < NJ;     k += BLK) async_ld_dw(&s_par[k], par,  (unsigned)(k * 4));
  if (tid < 3) async_ld_dw(&s_root[tid], rootp, (unsigned)(tid * 4));
  if (tid < NJ) s_done[tid] = 0;
  wait_async0();
  __syncthreads();

  const bool active = (tid < NJ);
  int  p    = active ? s_par[tid] : 0;
  bool done = !active;  // idle lanes are "done" from the start

  float wx = 0.f, wy = 0.f, wz = 0.f, ww = 0.f;
  float ox = 0.f, oy = 0.f, oz = 0.f;
  if (active) {
    wx = s_w[tid * 4 + 0]; wy = s_w[tid * 4 + 1];
    wz = s_w[tid * 4 + 2]; ww = s_w[tid * 4 + 3];
    ox = s_off[tid * 3 + 0]; oy = s_off[tid * 3 + 1]; oz = s_off[tid * 3 + 2];
  }

  // -- level-parallel tree resolution: ~e*ln(NJ) ~= 11 rounds expected ------
  // Two-phase per round: (read flags) barrier (compute+write) barrier,
  // so parent ori/pos reads are always separated from their writes.
  for (int round = 0; round < NJ + 1; ++round) {
    bool ready = false;
    if (!done) ready = (p < 0) || (s_done[p] != 0);
    __syncthreads();

    if (ready) {
      if (p < 0) {
        // root: ori = weight (raw), pos = root_position
        s_ori[tid * 4 + 0] = wx; s_ori[tid * 4 + 1] = wy;
        s_ori[tid * 4 + 2] = wz; s_ori[tid * 4 + 3] = ww;
        s_pos[tid * 3 + 0] = s_root[0];
        s_pos[tid * 3 + 1] = s_root[1];
        s_pos[tid * 3 + 2] = s_root[2];
      } else {
        const float px  = s_ori[p * 4 + 0], py = s_ori[p * 4 + 1];
        const float pz  = s_ori[p * 4 + 2], pw = s_ori[p * 4 + 3];
        const float ppx = s_pos[p * 3 + 0], ppy = s_pos[p * 3 + 1];
        const float ppz = s_pos[p * 3 + 2];

        // normalize parent quat: q/(||q|| + eps)
        const float pr = 1.0f / (sqrtf(px*px + py*py + pz*pz + pw*pw) + FK_EPS);
        const float x1 = px * pr, y1 = py * pr, z1 = pz * pr, w1 = pw * pr;
        // normalize my local quat
        const float wr = 1.0f / (sqrtf(wx*wx + wy*wy + wz*wz + ww*ww) + FK_EPS);
        const float x2 = wx * wr, y2 = wy * wr, z2 = wz * wr, w2 = ww * wr;

        // quat multiply (x,y,z,w layout, matches reference)
        const float nx = w1*x2 + x1*w2 + y1*z2 - z1*y2;
        const float ny = w1*y2 - x1*z2 + y1*w2 + z1*x2;
        const float nz = w1*z2 + x1*y2 - y1*x2 + z1*w2;
        const float nw = w1*w2 - x1*x2 - y1*y2 - z1*z2;

        // rotate offset by normalized parent quat
        const float xx = x1*x1, yy = y1*y1, zz = z1*z1, wwq = w1*w1;
        const float xy = x1*y1, zw = z1*w1, xz = x1*z1;
        const float yw = y1*w1, yz = y1*z1, xw = x1*w1;
        const float rx = (xx - yy - zz + wwq) * ox + 2.f*(xy - zw) * oy + 2.f*(xz + yw) * oz;
        const float ry = 2.f*(xy + zw) * ox + (-xx + yy - zz + wwq) * oy + 2.f*(yz - xw) * oz;
        const float rz = 2.f*(xz - yw) * ox + 2.f*(yz + xw) * oy + (-xx - yy + zz + wwq) * oz;

        s_ori[tid * 4 + 0] = nx; s_ori[tid * 4 + 1] = ny;
        s_ori[tid * 4 + 2] = nz; s_ori[tid * 4 + 3] = nw;
        s_pos[tid * 3 + 0] = rx + ppx;
        s_pos[tid * 3 + 1] = ry + ppy;
        s_pos[tid * 3 + 2] = rz + ppz;
      }
      s_done[tid] = 1;
      done = true;
    }

    if (__syncthreads_and(done ? 1 : 0)) break;  // barrier + early exit
  }

  // -- write result via async LDS->global store (ASYNCcnt path) -------------
  for (int k = tid; k < NJ * 3; k += BLK)
    async_st_dw(out, &s_pos[k], (unsigned)(k * 4));
  wait_async0();
}

// ---- host entry ------------------------------------------------------------
extern "C" void kernel_launch(void* const* d_in, const int* in_sizes, int n_in,
                              void* d_out, int out_size, void* d_ws, size_t ws_size,
                              hipStream_t stream) {
  (void)in_sizes; (void)n_in; (void)d_ws; (void)ws_size; (void)out_size;
  const float* joff  = (const float*)d_in[0];   // joint_offsets [65,3]
  const float* rootp = (const float*)d_in[1];   // root_position [3]
  const float* wgt   = (const float*)d_in[2];   // weight        [65,4]
  const int*   par   = (const int*)  d_in[3];   // joint_parents [65]
  float* out = (float*)d_out;                   // pos           [65,3]

  ForwardKinematicQuat_kernel<<<1, BLK, 0, stream>>>(joff, rootp, wgt, par, out);
}